// SelectTopKExperts_68710886801477
// MI455X (gfx1250) — compile-verified
//
#include <hip/hip_runtime.h>
#include <hip/hip_bf16.h>
#include <math.h>

#define TOPK 8
#define NEXP 256          // experts per row (fixed by problem)
#define PER_LANE 8        // NEXP / 32 lanes

// One wave32 per batch row.
//  - lane l loads logits[row, l*8 .. l*8+7] as a single 16B global_load_b128
//  - keys: (monotone16(bf16) << 16) | (255 - expert_idx)
//      -> u32 max == "max value, ties -> lowest index"  (jax.lax.top_k order)
//  - 8 selection rounds, each a per-lane max + 5-step __shfl_xor butterfly
//  - weights: exp(v - vmax) / sum over the 8 selected (softmax denom cancels)
//  - lanes 0..7 do the EPLB dispatch lookup and write outputs
__global__ __launch_bounds__(256, 4) void
moe_topk_router_kernel(const __hip_bfloat16* __restrict__ logits,   // [B, 256]
                       const int*            __restrict__ l2p,      // [256, P]
                       const int*            __restrict__ num_valid,// [256]
                       const int*            __restrict__ drand,    // [B, 8]
                       float*                __restrict__ out,      // weights @0, ids @B*8
                       int batch, int phys_stride, int ids64)
{
    const int lane = threadIdx.x & 31;
    const int wave = threadIdx.x >> 5;
    const int row  = blockIdx.x * (blockDim.x >> 5) + wave;
    if (row >= batch) return;

    // ---- coalesced 16B-per-lane load of this row's 256 bf16 logits ----
    const unsigned short* rowbase =
        (const unsigned short*)logits + (size_t)row * NEXP;
    const uint4 v = *((const uint4*)rowbase + lane);   // global_load_b128

    unsigned short e[PER_LANE];
    e[0] = (unsigned short)(v.x & 0xFFFFu); e[1] = (unsigned short)(v.x >> 16);
    e[2] = (unsigned short)(v.y & 0xFFFFu); e[3] = (unsigned short)(v.y >> 16);
    e[4] = (unsigned short)(v.z & 0xFFFFu); e[5] = (unsigned short)(v.z >> 16);
    e[6] = (unsigned short)(v.w & 0xFFFFu); e[7] = (unsigned short)(v.w >> 16);

    // ---- order-preserving pack: value(16b) | (255 - index)(8b) ----
    unsigned int key[PER_LANE];
#pragma unroll
    for (int i = 0; i < PER_LANE; ++i) {
        const unsigned short b = e[i];
        const unsigned short m = (b & 0x8000u) ? (unsigned short)(~b)
                                               : (unsigned short)(b | 0x8000u);
        const int idx = lane * PER_LANE + i;
        key[i] = ((unsigned int)m << 16) | (unsigned int)(255 - idx);
    }

    // ---- iterative top-8: wave32 butterfly max-reduce, then consume winner ----
    unsigned int kept[TOPK];
#pragma unroll
    for (int k = 0; k < TOPK; ++k) {
        unsigned int best = key[0];
#pragma unroll
        for (int i = 1; i < PER_LANE; ++i) best = (key[i] > best) ? key[i] : best;
#pragma unroll
        for (int off = 16; off > 0; off >>= 1) {
            unsigned int o = (unsigned int)__shfl_xor((int)best, off, 32);
            best = (o > best) ? o : best;
        }
        kept[k] = best;                                 // wave-uniform
        const int widx = 255 - (int)(best & 0xFFu);     // winning expert index
        if ((widx >> 3) == lane) key[widx & 7] = 0u;    // mark consumed
    }

    // ---- decode values, softmax over the selected 8 (denominator cancels) ----
    float f[TOPK];
    float vmax;
    {
        const unsigned int m0 = kept[0] >> 16;
        const unsigned short b0 = (m0 & 0x8000u) ? (unsigned short)(m0 ^ 0x8000u)
                                                 : (unsigned short)(~m0 & 0xFFFFu);
        vmax = __uint_as_float((unsigned int)b0 << 16);
    }
    float s = 0.0f;
#pragma unroll
    for (int k = 0; k < TOPK; ++k) {
        const unsigned int m = kept[k] >> 16;
        const unsigned short b = (m & 0x8000u) ? (unsigned short)(m ^ 0x8000u)
                                               : (unsigned short)(~m & 0xFFFFu);
        const float val = __uint_as_float((unsigned int)b << 16);
        f[k] = expf(val - vmax);
        s += f[k];
    }

    // ---- lanes 0..7: EPLB dispatch + output ----
    if (lane < TOPK) {
        const int k  = lane;
        const int id = 255 - (int)(kept[k] & 0xFFu);
        const int nv = num_valid[id];
        const int r  = drand[(size_t)row * TOPK + k];
        const int chosen = r % nv;
        const int phys = l2p[(size_t)id * phys_stride + chosen];

        const size_t o = (size_t)row * TOPK + k;
        out[o] = f[k] / s;                               // renormalized weight
        if (ids64) {
            // out_size implied 64-bit id storage after the fp32 weights
            long long* ids = (long long*)((char*)out + (size_t)batch * TOPK * 4);
            ids[o] = (long long)phys;
        } else {
            out[(size_t)batch * TOPK + o] = (float)phys; // ids as f32 (0..287 exact)
        }
    }
}

extern "C" void kernel_launch(void* const* d_in, const int* in_sizes, int n_in,
                              void* d_out, int out_size, void* d_ws, size_t ws_size,
                              hipStream_t stream) {
    // setup_inputs order:
    //  [0] hidden_states  (bf16, UNUSED by the reference output)
    //  [1] router_logits  (bf16, [B,256])
    //  [2] logical_to_physical_map (int, [256,P])
    //  [3] num_valid      (int, [256])
    //  [4] dispatch_rand  (int32, [B,8])
    const __hip_bfloat16* logits    = (const __hip_bfloat16*)d_in[1];
    const int*            l2p       = (const int*)d_in[2];
    const int*            num_valid = (const int*)d_in[3];
    const int*            drand     = (const int*)d_in[4];

    const int batch       = in_sizes[4] / TOPK;
    const int nexp        = in_sizes[3];
    const int phys_stride = in_sizes[2] / (nexp > 0 ? nexp : 1);
    // out is weights(f32) ++ ids; if out_size suggests 8-byte ids, write int64.
    const int ids64 = (out_size == batch * TOPK * 3) ? 1 : 0;

    const int waves_per_block = 8;                   // 256 threads
    dim3 block(32 * waves_per_block);
    dim3 grid((batch + waves_per_block - 1) / waves_per_block);
    moe_topk_router_kernel<<<grid, block, 0, stream>>>(
        logits, l2p, num_valid, drand, (float*)d_out, batch, phys_stride, ids64);
}